// HeteroGNN_26912265077247
// MI455X (gfx1250) — compile-verified
//
#include <hip/hip_runtime.h>

// ---------------- problem constants ----------------
#define N_NODES 100000
#define N_EDGES 800000
#define NT 3
#define NL 2
#define NH 2
#define NC 64
#define NHC 128   // H*C

#define ROWTILES 8          // 16-row tiles per block (128 rows/block)
#define LDS_PITCH 68        // 64 + 4 pad: 16B-aligned rows, conflict-free frags

typedef float v2f __attribute__((ext_vector_type(2)));
typedef float v8f __attribute__((ext_vector_type(8)));

// ---------------- atomic helpers (native CDNA5 paths) ----------------
__device__ __forceinline__ void atomicMaxF(float* addr, float v) {
    // float max via integer atomics: positives compare as signed int,
    // negatives compare reversed as unsigned. Init value must be -inf.
    if (v >= 0.0f) atomicMax((int*)addr, __float_as_int(v));
    else           atomicMin((unsigned int*)addr, __float_as_uint(v));
}

__device__ __forceinline__ void atomAddF(float* addr, float v) {
    // maps to global_atomic_add_f32 (L2 atomic unit), no CAS loop
    unsafeAtomicAdd(addr, v);
}

// ---------------- init: zero agg + seg_sum, -inf seg_max ----------------
__global__ void init_k(float* __restrict__ agg, float* __restrict__ ssum,
                       float* __restrict__ smax) {
    int idx = blockIdx.x * blockDim.x + threadIdx.x;   // N*HC threads exactly
    agg[idx] = 0.0f;
    if (idx < NT * N_NODES * NH) {
        ssum[idx] = 0.0f;
        smax[idx] = -__builtin_inff();
    }
}

// ---------------- fp32 WMMA GEMM with B-register blocking + LDS A staging ---
// out[z] = x @ W[z]  (+ bias).  NCOLS in {64,128}; blockDim.x = 2*NCOLS
// (NCOLS/16 waves, one 16-col tile per wave). Each block covers ROWTILES
// 16-row tiles: B fragments live in registers across all row tiles, the A
// tile is staged once in LDS and shared by all waves.
template<int NCOLS, bool HAS_BIAS>
__global__ void gemm_wmma_k(const float* __restrict__ x,
                            const float* __restrict__ Wall,
                            const float* __restrict__ bias,
                            float* __restrict__ outAll) {
    __shared__ float As[16 * LDS_PITCH];

    const float* W  = Wall   + (size_t)blockIdx.z * NC * NCOLS;
    float*      out = outAll + (size_t)blockIdx.z * N_NODES * NCOLS;

    const int lane = threadIdx.x & 31;
    const int wave = threadIdx.x >> 5;
    const int col0 = wave * 16;
    const int m    = lane & 15;
    const int kk   = (lane >> 4) << 1;          // 0 or 2: fragment K pair

    // --- preload all 16 B fragments for this wave's col tile (32 VGPRs) ---
    v2f bfr[16];
#pragma unroll
    for (int i = 0; i < 16; ++i) {
        const int k0 = i * 4;
        bfr[i].x = W[(k0 + kk)     * NCOLS + col0 + m];
        bfr[i].y = W[(k0 + kk + 1) * NCOLS + col0 + m];
    }
    const float bv = HAS_BIAS ? bias[col0 + m] : 0.0f;

    const int rowBlock = blockIdx.x * (ROWTILES * 16);
#pragma unroll 1
    for (int rt = 0; rt < ROWTILES; ++rt) {
        const int r0 = rowBlock + rt * 16;
        if (r0 >= N_NODES) break;               // block-uniform: EXEC stays full

        // --- coalesced cooperative stage of the 16x64 A tile into LDS ---
        __syncthreads();
        const float4* src4 = (const float4*)(x + (size_t)r0 * NC);
        float4*       dst4 = (float4*)As;
        for (int i = threadIdx.x; i < 16 * (NC / 4); i += blockDim.x) {
            const int row = i >> 4, j = i & 15;          // 16 float4 per row
            dst4[row * (LDS_PITCH / 4) + j] = src4[i];
        }
        __syncthreads();

        // --- 16 chained fp32 WMMAs over K=64 ---
        v8f acc = {};
#pragma unroll
        for (int i = 0; i < 16; ++i) {
            const int k0 = i * 4;
            v2f a = *(const v2f*)(As + m * LDS_PITCH + k0 + kk);  // ds_load_b64
            acc = __builtin_amdgcn_wmma_f32_16x16x4_f32(
                false, a, false, bfr[i], (short)0, acc, false, false);
        }

        // --- store C tile (row-major [M=16][N=16] striping per ISA) ---
        const int rbase = r0 + ((lane >> 4) << 3);
        const int col   = col0 + m;
#pragma unroll
        for (int r = 0; r < 8; ++r)
            out[(size_t)(rbase + r) * NCOLS + col] = acc[r] + bv;
    }
}

// ---------------- attention logits per node: al_s/al_d [T,N,H] ----------------
__global__ void alpha_k(const float* __restrict__ hbuf,
                        const float* __restrict__ asrc,
                        const float* __restrict__ adst,
                        float* __restrict__ als, float* __restrict__ ald) {
    int idx = blockIdx.x * blockDim.x + threadIdx.x;   // t*(N*H)+n*H+h
    if (idx >= NT * N_NODES * NH) return;
    int h = idx & 1;
    int n = (idx >> 1) % N_NODES;
    int t = idx / (N_NODES * NH);
    const float4* hp = (const float4*)(hbuf + ((size_t)t * N_NODES + n) * NHC + h * NC);
    const float4* as = (const float4*)(asrc + (size_t)(t * NH + h) * NC);
    const float4* ad = (const float4*)(adst + (size_t)(t * NH + h) * NC);
    float s = 0.0f, d = 0.0f;
#pragma unroll
    for (int i = 0; i < 16; ++i) {
        float4 hv = hp[i], av = as[i], bv = ad[i];
        s += hv.x * av.x + hv.y * av.y + hv.z * av.z + hv.w * av.w;
        d += hv.x * bv.x + hv.y * bv.y + hv.z * bv.z + hv.w * bv.w;
    }
    als[idx] = s;
    ald[idx] = d;
}

// ---------------- edge pass 1: leaky-relu logits + segment max ----------------
__global__ void edge_logit_k(const int* __restrict__ ei,
                             const float* __restrict__ als,
                             const float* __restrict__ ald,
                             float* __restrict__ pbuf,
                             float* __restrict__ smax) {
    int idx = blockIdx.x * blockDim.x + threadIdx.x;   // t*E+e
    if (idx >= NT * N_EDGES) return;
    int t = idx / N_EDGES;
    int e = idx - t * N_EDGES;
    int src = ei[(size_t)t * 2 * N_EDGES + e];
    int dst = ei[(size_t)t * 2 * N_EDGES + N_EDGES + e];
    float2 s2 = *(const float2*)(als + ((size_t)t * N_NODES + src) * 2);
    float2 d2 = *(const float2*)(ald + ((size_t)t * N_NODES + dst) * 2);
    float l0 = s2.x + d2.x; l0 = l0 > 0.0f ? l0 : 0.2f * l0;
    float l1 = s2.y + d2.y; l1 = l1 > 0.0f ? l1 : 0.2f * l1;
    pbuf[(size_t)idx * 2 + 0] = l0;
    pbuf[(size_t)idx * 2 + 1] = l1;
    float* mx = smax + ((size_t)t * N_NODES + dst) * 2;
    atomicMaxF(mx + 0, l0);
    atomicMaxF(mx + 1, l1);
}

// ---------------- edge pass 2: exp + segment sum ----------------
__global__ void edge_exp_k(const int* __restrict__ ei,
                           float* __restrict__ pbuf,
                           const float* __restrict__ smax,
                           float* __restrict__ ssum) {
    int idx = blockIdx.x * blockDim.x + threadIdx.x;
    if (idx >= NT * N_EDGES) return;
    int t = idx / N_EDGES;
    int dst = ei[(size_t)t * 2 * N_EDGES + N_EDGES + (idx - t * N_EDGES)];
    float2 m = *(const float2*)(smax + ((size_t)t * N_NODES + dst) * 2);
    float p0 = __expf(pbuf[(size_t)idx * 2 + 0] - m.x);
    float p1 = __expf(pbuf[(size_t)idx * 2 + 1] - m.y);
    pbuf[(size_t)idx * 2 + 0] = p0;
    pbuf[(size_t)idx * 2 + 1] = p1;
    float* s = ssum + ((size_t)t * N_NODES + dst) * 2;
    atomAddF(s + 0, p0);
    atomAddF(s + 1, p1);
}

// ---------------- edge pass 3: message scatter, one wave32 per edge ----------
// lane handles 4 contiguous channels; lanes 0-15 head 0, 16-31 head 1.
__global__ void edge_msg_k(const int* __restrict__ ei,
                           const float* __restrict__ hbuf,
                           const float* __restrict__ pbuf,
                           const float* __restrict__ ssum,
                           float* __restrict__ agg) {
    long gid = (long)blockIdx.x * blockDim.x + threadIdx.x;  // T*E*32 exactly
    int  lane = (int)(gid & 31);
    long w = gid >> 5;                                       // edge id over T*E
    int t = (int)(w / N_EDGES);
    int e = (int)(w - (long)t * N_EDGES);
    int src = ei[(size_t)t * 2 * N_EDGES + e];
    int dst = ei[(size_t)t * 2 * N_EDGES + N_EDGES + e];
    int head = lane >> 4;
    float alpha = pbuf[(size_t)w * 2 + head] /
                  ssum[((size_t)t * N_NODES + dst) * 2 + head];
    float4 hv = *(const float4*)(hbuf + ((size_t)t * N_NODES + src) * NHC + lane * 4);
    float* ap = agg + (size_t)dst * NHC + lane * 4;
    atomAddF(ap + 0, hv.x * alpha);
    atomAddF(ap + 1, hv.y * alpha);
    atomAddF(ap + 2, hv.z * alpha);
    atomAddF(ap + 3, hv.w * alpha);
}

// -------- epilogue: x' = relu(3*shared + mean_h(agg) + sum_t bias_t) --------
__global__ void combine_k(const float* __restrict__ shared,
                          const float* __restrict__ agg,
                          const float* __restrict__ bias,   // [T,C] for this layer
                          float* __restrict__ xout) {
    int idx = blockIdx.x * blockDim.x + threadIdx.x;   // N*C exactly
    int c = idx & (NC - 1);
    int n = idx >> 6;
    float bsum = bias[c] + bias[NC + c] + bias[2 * NC + c];
    float v = 3.0f * shared[idx]
            + 0.5f * (agg[(size_t)n * NHC + c] + agg[(size_t)n * NHC + NC + c])
            + bsum;
    xout[idx] = v > 0.0f ? v : 0.0f;
}

// ---------------- host orchestration ----------------
extern "C" void kernel_launch(void* const* d_in, const int* in_sizes, int n_in,
                              void* d_out, int out_size, void* d_ws, size_t ws_size,
                              hipStream_t stream) {
    const float* x_in     = (const float*)d_in[0];
    const float* W_sh     = (const float*)d_in[1];
    const float* b_sh     = (const float*)d_in[2];
    const float* W_gat    = (const float*)d_in[3];
    const float* att_src  = (const float*)d_in[4];
    const float* att_dst  = (const float*)d_in[5];
    const float* bias_gat = (const float*)d_in[6];
    const int*   ei       = (const int*)d_in[7];
    float* out = (float*)d_out;

    float* ws = (float*)d_ws;
    float* x_cur  = ws; ws += (size_t)N_NODES * NC;
    float* shared = ws; ws += (size_t)N_NODES * NC;
    float* hbuf   = ws; ws += (size_t)NT * N_NODES * NHC;
    float* al_s   = ws; ws += (size_t)NT * N_NODES * NH;
    float* al_d   = ws; ws += (size_t)NT * N_NODES * NH;
    float* smax   = ws; ws += (size_t)NT * N_NODES * NH;
    float* ssum   = ws; ws += (size_t)NT * N_NODES * NH;
    float* pbuf   = ws; ws += (size_t)NT * N_EDGES * NH;
    float* agg    = ws; ws += (size_t)N_NODES * NHC;

    const int rowBlocks = (N_NODES / 16 + ROWTILES - 1) / ROWTILES;  // 782

    for (int l = 0; l < NL; ++l) {
        const float* xin = (l == 0) ? x_in : x_cur;
        float* xout = (l == NL - 1) ? out : x_cur;

        init_k<<<(N_NODES * NHC) / 256, 256, 0, stream>>>(agg, ssum, smax);

        // shared linear: 64 cols -> 4 waves (128 threads)
        gemm_wmma_k<NC, true><<<dim3(rowBlocks, 1, 1), 128, 0, stream>>>(
            xin, W_sh + (size_t)l * NC * NC, b_sh + (size_t)l * NC, shared);

        // per-type GAT linears: 128 cols -> 8 waves (256 threads), z = type
        gemm_wmma_k<NHC, false><<<dim3(rowBlocks, 1, NT), 256, 0, stream>>>(
            xin, W_gat + (size_t)l * NT * NC * NHC, nullptr, hbuf);

        alpha_k<<<(NT * N_NODES * NH + 255) / 256, 256, 0, stream>>>(
            hbuf, att_src + (size_t)l * NT * NH * NC,
            att_dst + (size_t)l * NT * NH * NC, al_s, al_d);

        edge_logit_k<<<(NT * N_EDGES) / 256, 256, 0, stream>>>(
            ei, al_s, al_d, pbuf, smax);

        edge_exp_k<<<(NT * N_EDGES) / 256, 256, 0, stream>>>(
            ei, pbuf, smax, ssum);

        edge_msg_k<<<((size_t)NT * N_EDGES * 32) / 256, 256, 0, stream>>>(
            ei, hbuf, pbuf, ssum, agg);

        combine_k<<<(N_NODES * NC) / 256, 256, 0, stream>>>(
            shared, agg, bias_gat + (size_t)l * NT * NC, xout);
    }
}